// SelfMultiheadAttention_25666724561251
// MI455X (gfx1250) — compile-verified
//
#include <hip/hip_runtime.h>
#include <math.h>

// Problem constants (match reference)
constexpr int N  = 8192;     // nodes
constexpr int E  = 256;      // embed dim
constexpr int H  = 8;        // heads
constexpr int D  = 32;       // head dim
constexpr int NE = 262144;   // edges
constexpr int EC = E + D;    // combined GEMM output channels: 256 q + 32 ksum
constexpr float SCAL = 0.17677669529663687f;  // 1/sqrt(D)

typedef float v2f __attribute__((ext_vector_type(2)));
typedef float v4f __attribute__((ext_vector_type(4)));
typedef float v8f __attribute__((ext_vector_type(8)));

// ---------------------------------------------------------------------------
// Prep: fold scaling into W_q, fold head-sum into W_k -> Wc [EC=288, 256], bc
// Wc rows 0..255   = SCAL * W_in[c]           (q path, scaling pre-folded)
// Wc rows 256..287 = sum_h W_in[256 + h*32+d] (k head-sum fold)
// ---------------------------------------------------------------------------
__global__ __launch_bounds__(256) void prep_kernel(
    const float* __restrict__ W_in, const float* __restrict__ b_in,
    float* __restrict__ Wc, float* __restrict__ bc) {
  int c = blockIdx.x;          // 0..287
  int k = threadIdx.x;         // 0..255
  if (c < E) {
    Wc[(size_t)c * E + k] = SCAL * W_in[(size_t)c * E + k];
    if (k == 0) bc[c] = SCAL * b_in[c];
  } else {
    int d = c - E;
    float s = 0.f;
    #pragma unroll
    for (int h = 0; h < H; ++h) s += W_in[(size_t)(E + h * D + d) * E + k];
    Wc[(size_t)c * E + k] = s;
    if (k == 0) {
      float sb = 0.f;
      #pragma unroll
      for (int h = 0; h < H; ++h) sb += b_in[E + h * D + d];
      bc[c] = sb;
    }
  }
}

// ---------------------------------------------------------------------------
// GEMM via V_WMMA_F32_16X16X4_F32:  out[n,c] = query[n,:] . Wc[c,:] + bc[c]
// One wave (32 threads) computes one 16x16 tile, K-loop step 8 (2 WMMAs/iter).
// A layout (16x4 f32): lane l -> M=l%16, vgpr j holds K=2*(l/16)+j
// B layout (4x16 f32): lane l -> N=l%16, vgpr j holds K=2*(l/16)+j
// K-label <-> global-k mapping identical for A and B (float4 split lo/hi).
// C/D layout: vgpr r -> M = r + 8*(l/16), N = l%16.
// Columns <256 -> qs (ld=256, == d_out scratch); columns >=256 -> ksum (ld=32)
// ---------------------------------------------------------------------------
__global__ __launch_bounds__(32) void gemm_qk_kernel(
    const float* __restrict__ query, const float* __restrict__ Wc,
    const float* __restrict__ bc, float* __restrict__ qs,
    float* __restrict__ ksum) {
  const int lane  = threadIdx.x;
  const int lm    = lane & 15;   // M (for A) / N (for B) within tile
  const int lh    = lane >> 4;   // half-wave -> K sub-block
  const int nbase = blockIdx.x * 16;   // 0..287
  const int mbase = blockIdx.y * 16;   // 0..8191

  const float* arow = query + (size_t)(mbase + lm) * E + 4 * lh;
  const float* brow = Wc    + (size_t)(nbase + lm) * E + 4 * lh;

  v8f acc = {0.f, 0.f, 0.f, 0.f, 0.f, 0.f, 0.f, 0.f};

  #pragma unroll 8
  for (int kb = 0; kb < E; kb += 8) {
    v4f a4 = *reinterpret_cast<const v4f*>(arow + kb);
    v4f b4 = *reinterpret_cast<const v4f*>(brow + kb);
    v2f aLo = __builtin_shufflevector(a4, a4, 0, 1);
    v2f aHi = __builtin_shufflevector(a4, a4, 2, 3);
    v2f bLo = __builtin_shufflevector(b4, b4, 0, 1);
    v2f bHi = __builtin_shufflevector(b4, b4, 2, 3);
    acc = __builtin_amdgcn_wmma_f32_16x16x4_f32(false, aLo, false, bLo,
                                                (short)0, acc, false, false);
    acc = __builtin_amdgcn_wmma_f32_16x16x4_f32(false, aHi, false, bHi,
                                                (short)0, acc, false, false);
  }

  const float bias = bc[nbase + lm];
  if (nbase < E) {
    float* base = qs + nbase + lm;
    #pragma unroll
    for (int r = 0; r < 8; ++r) {
      int m = r + 8 * lh;
      base[(size_t)(mbase + m) * E] = acc[r] + bias;
    }
  } else {
    float* base = ksum + (nbase - E) + lm;
    #pragma unroll
    for (int r = 0; r < 8; ++r) {
      int m = r + 8 * lh;
      base[(size_t)(mbase + m) * D] = acc[r] + bias;
    }
  }
}

// ---------------------------------------------------------------------------
// Edge phase: one thread per (edge, head).
// s = q[dst, h, :] . ksum[src, :] + attn_bias[h, e];  col[h, src] += s
// ---------------------------------------------------------------------------
__global__ __launch_bounds__(256) void edge_kernel(
    const int* __restrict__ ei, const float* __restrict__ attn_bias,
    const float* __restrict__ qs, const float* __restrict__ ksum,
    float* __restrict__ col, unsigned* __restrict__ touched) {
  int tid = blockIdx.x * 256 + threadIdx.x;
  int e = tid >> 3;
  int h = tid & 7;
  if (e >= NE) return;
  int src = ei[e];
  int dst = ei[NE + e];
  const v4f* q4 = reinterpret_cast<const v4f*>(qs + (size_t)dst * E + h * D);
  const v4f* k4 = reinterpret_cast<const v4f*>(ksum + (size_t)src * D);
  float s = attn_bias[(size_t)h * NE + e];
  #pragma unroll
  for (int i = 0; i < 8; ++i) {
    v4f a = q4[i], b = k4[i];
    s += a.x * b.x + a.y * b.y + a.z * b.z + a.w * b.w;
  }
  atomicAdd(col + (size_t)h * N + src, s);
  if (h == 0) touched[src] = 1u;
}

// ---------------------------------------------------------------------------
// Softmax over nodes per head (untouched -> -inf -> weight 0).
// One block (8 waves) per head.
// ---------------------------------------------------------------------------
__global__ __launch_bounds__(256) void softmax_kernel(
    const float* __restrict__ col, const unsigned* __restrict__ touched,
    float* __restrict__ attn) {
  const int h = blockIdx.x;
  const int tid = threadIdx.x;
  const float* c = col + (size_t)h * N;
  __shared__ float red[8];
  __shared__ float bM, bS;

  float m = -INFINITY;
  for (int n = tid; n < N; n += 256) {
    float cv = touched[n] ? c[n] : -INFINITY;
    m = fmaxf(m, cv);
  }
  #pragma unroll
  for (int o = 16; o > 0; o >>= 1) m = fmaxf(m, __shfl_xor(m, o, 32));
  if ((tid & 31) == 0) red[tid >> 5] = m;
  __syncthreads();
  if (tid == 0) {
    float mm = red[0];
    #pragma unroll
    for (int i = 1; i < 8; ++i) mm = fmaxf(mm, red[i]);
    bM = mm;
  }
  __syncthreads();
  const float M = bM;

  float s = 0.f;
  for (int n = tid; n < N; n += 256) {
    if (touched[n]) s += __expf(c[n] - M);
  }
  #pragma unroll
  for (int o = 16; o > 0; o >>= 1) s += __shfl_xor(s, o, 32);
  if ((tid & 31) == 0) red[tid >> 5] = s;
  __syncthreads();
  if (tid == 0) {
    float ss = 0.f;
    #pragma unroll
    for (int i = 0; i < 8; ++i) ss += red[i];
    bS = ss;
  }
  __syncthreads();
  const float inv = 1.0f / bS;

  for (int n = tid; n < N; n += 256) {
    attn[(size_t)h * N + n] = touched[n] ? __expf(c[n] - M) * inv : 0.f;
  }
}

// ---------------------------------------------------------------------------
// r[h, :] = sum_n attn[h,n] * query[n, :]    ([8,256], atomic-accumulated)
// ---------------------------------------------------------------------------
__global__ __launch_bounds__(256) void r_kernel(
    const float* __restrict__ attn, const float* __restrict__ query,
    float* __restrict__ r) {
  const int h = blockIdx.y;
  const int n0 = blockIdx.x * 256;
  const int cidx = threadIdx.x;
  float acc = 0.f;
  for (int i = 0; i < 256; ++i) {
    float a = attn[(size_t)h * N + n0 + i];
    acc += a * query[(size_t)(n0 + i) * E + cidx];
  }
  atomicAdd(&r[h * E + cidx], acc);
}

// ---------------------------------------------------------------------------
// z[c] = r[h(c), :] . W_v[c, :] + b_v[c]   (W_v row c = W_in row 2E+c)
// ---------------------------------------------------------------------------
__global__ __launch_bounds__(256) void z_kernel(
    const float* __restrict__ r, const float* __restrict__ W_in,
    const float* __restrict__ b_in, float* __restrict__ z) {
  const int c = threadIdx.x;
  const int h = c >> 5;
  const float* w = W_in + (size_t)(2 * E + c) * E;
  const float* rr = r + h * E;
  float acc = b_in[2 * E + c];
  for (int k = 0; k < E; ++k) acc += rr[k] * w[k];
  z[c] = acc;
}

// ---------------------------------------------------------------------------
// y[j] = W_out[j,:] . z + b_out[j]
// ---------------------------------------------------------------------------
__global__ __launch_bounds__(256) void y_kernel(
    const float* __restrict__ z, const float* __restrict__ W_out,
    const float* __restrict__ b_out, float* __restrict__ y) {
  const int j = threadIdx.x;
  const float* w = W_out + (size_t)j * E;
  float acc = b_out[j];
  for (int c = 0; c < E; ++c) acc += w[c] * z[c];
  y[j] = acc;
}

// Broadcast the single output row to all N rows.
__global__ __launch_bounds__(256) void bcast_kernel(
    const float* __restrict__ y, float* __restrict__ out) {
  out[(size_t)blockIdx.x * 256 + threadIdx.x] = y[threadIdx.x];
}

// ---------------------------------------------------------------------------
// Workspace layout (in floats)
// ---------------------------------------------------------------------------
constexpr size_t WC_OFF  = 0;                               // EC*E = 73728
constexpr size_t BC_OFF  = WC_OFF + (size_t)EC * E;         // 512 (padded)
constexpr size_t KS_OFF  = BC_OFF + 512;                    // N*D
constexpr size_t COL_OFF = KS_OFF + (size_t)N * D;          // H*N
constexpr size_t ATT_OFF = COL_OFF + (size_t)H * N;         // H*N
constexpr size_t TCH_OFF = ATT_OFF + (size_t)H * N;         // N (uint)
constexpr size_t R_OFF   = TCH_OFF + (size_t)N;             // H*E
constexpr size_t Z_OFF   = R_OFF + (size_t)H * E;           // E
constexpr size_t Y_OFF   = Z_OFF + (size_t)E;               // E

extern "C" void kernel_launch(void* const* d_in, const int* in_sizes, int n_in,
                              void* d_out, int out_size, void* d_ws, size_t ws_size,
                              hipStream_t stream) {
  const float* query     = (const float*)d_in[0];
  const int*   edgeindex = (const int*)d_in[1];
  const float* attn_bias = (const float*)d_in[2];
  const float* W_in      = (const float*)d_in[3];
  const float* b_in      = (const float*)d_in[4];
  const float* W_out     = (const float*)d_in[5];
  const float* b_out     = (const float*)d_in[6];

  float* ws = (float*)d_ws;
  float* Wc      = ws + WC_OFF;
  float* bc      = ws + BC_OFF;
  float* ksum    = ws + KS_OFF;
  float* col     = ws + COL_OFF;
  float* attn    = ws + ATT_OFF;
  unsigned* tch  = (unsigned*)(ws + TCH_OFF);
  float* r       = ws + R_OFF;
  float* z       = ws + Z_OFF;
  float* y       = ws + Y_OFF;

  float* qs = (float*)d_out;  // reuse output buffer as q scratch ([N,E] f32)

  hipMemsetAsync(col, 0, (size_t)H * N * sizeof(float), stream);
  hipMemsetAsync(tch, 0, (size_t)N * sizeof(unsigned), stream);
  hipMemsetAsync(r, 0, (size_t)H * E * sizeof(float), stream);

  prep_kernel<<<EC, 256, 0, stream>>>(W_in, b_in, Wc, bc);

  dim3 ggrid(EC / 16, N / 16);  // 18 x 512 tiles, one wave each
  gemm_qk_kernel<<<ggrid, 32, 0, stream>>>(query, Wc, bc, qs, ksum);

  edge_kernel<<<(NE * H) / 256, 256, 0, stream>>>(edgeindex, attn_bias, qs,
                                                  ksum, col, tch);

  softmax_kernel<<<H, 256, 0, stream>>>(col, tch, attn);

  dim3 rgrid(N / 256, H);
  r_kernel<<<rgrid, 256, 0, stream>>>(attn, query, r);

  z_kernel<<<1, 256, 0, stream>>>(r, W_in, b_in, z);
  y_kernel<<<1, 256, 0, stream>>>(z, W_out, b_out, y);

  bcast_kernel<<<N, 256, 0, stream>>>(y, (float*)d_out);
}